// Graph_Attention_43361989820507
// MI455X (gfx1250) — compile-verified
//
#include <hip/hip_runtime.h>
#include <math.h>

#define N_STOCK 198
#define F_IN    512
#define F_OUT   2048
#define M_TILES 13              // ceil(198/16)
#define NT      2               // n-tiles (16 cols each) per wave

typedef __attribute__((ext_vector_type(2))) float v2f;
typedef __attribute__((ext_vector_type(8))) float v8f;

// ---------------------------------------------------------------------------
// GEMM: out[m,n] = sum_k X[m,k] * W[n,k] (+ bias[n])
// X: [198,512] row-major, W: [2048,512] row-major, out: [198,2048]
// One wave (32 threads) computes a 16 x (16*NT) output tile with
// V_WMMA_F32_16X16X4_F32 accumulation over K=512.
//
// Fragment addressing for 16x16x4 f32 (per CDNA5 ISA):
//   A (16x4, MxK): lanes 0-15 hold row M=lane, VGPR0=K0/V1=K1;
//                  lanes 16-31 hold row M=lane-16, VGPR0=K2/V1=K3.
//   B (4x16, KxN): symmetric — lane half selects K pair, lane%16 selects N.
// => both A and B fragments are a single b64 load per lane:
//      row = base + (lane&15), col = k + 2*(lane>>4) .. +1
// ---------------------------------------------------------------------------
__global__ __launch_bounds__(32) void gemm_xwT_wmma(
    const float* __restrict__ X, const float* __restrict__ W,
    const float* __restrict__ bias, float* __restrict__ out)
{
    const int lane = threadIdx.x;
    const int half = lane >> 4;
    const int lr   = lane & 15;
    const int m0   = blockIdx.x * 16;
    const int n0   = blockIdx.y * (16 * NT);

    int mrow = m0 + lr;
    if (mrow >= N_STOCK) mrow = N_STOCK - 1;   // clamp; padded rows never stored

    const float* aptr  = X + (size_t)mrow * F_IN + 2 * half;
    const float* bptr0 = W + (size_t)(n0 +  0 + lr) * F_IN + 2 * half;
    const float* bptr1 = W + (size_t)(n0 + 16 + lr) * F_IN + 2 * half;

    v8f acc0 = {};
    v8f acc1 = {};

    #pragma unroll 4
    for (int k = 0; k < F_IN; k += 4) {
        v2f a  = *(const v2f*)(aptr  + k);
        v2f b0 = *(const v2f*)(bptr0 + k);
        v2f b1 = *(const v2f*)(bptr1 + k);
        acc0 = __builtin_amdgcn_wmma_f32_16x16x4_f32(false, a, false, b0,
                                                     (short)0, acc0, false, false);
        acc1 = __builtin_amdgcn_wmma_f32_16x16x4_f32(false, a, false, b1,
                                                     (short)0, acc1, false, false);
    }

    const float bv0 = bias ? bias[n0 +  0 + lr] : 0.0f;
    const float bv1 = bias ? bias[n0 + 16 + lr] : 0.0f;

    // C/D layout: VGPR v holds M = m0 + 8*half + v, N = n0 + (lane&15)
    #pragma unroll
    for (int v = 0; v < 8; ++v) {
        const int m = m0 + half * 8 + v;
        if (m < N_STOCK) {
            out[(size_t)m * F_OUT + n0 +  0 + lr] = acc0[v] + bv0;
            out[(size_t)m * F_OUT + n0 + 16 + lr] = acc1[v] + bv1;
        }
    }
}

// ---------------------------------------------------------------------------
// f1[i] = seq_r[i,:] . f1_w + f1_b ;  f2[i] = seq_r[i,:] . f2_w + f2_b
// one block per row
// ---------------------------------------------------------------------------
__global__ __launch_bounds__(256) void f1f2_kernel(
    const float* __restrict__ seq_r,
    const float* __restrict__ f1w, const float* __restrict__ f2w,
    const float* __restrict__ f1b, const float* __restrict__ f2b,
    float* __restrict__ f1, float* __restrict__ f2)
{
    __shared__ float s1[256];
    __shared__ float s2[256];
    const int i = blockIdx.x;
    const float* row = seq_r + (size_t)i * F_OUT;
    float a1 = 0.0f, a2 = 0.0f;
    for (int f = threadIdx.x; f < F_OUT; f += 256) {
        const float v = row[f];
        a1 += v * f1w[f];
        a2 += v * f2w[f];
    }
    s1[threadIdx.x] = a1;
    s2[threadIdx.x] = a2;
    __syncthreads();
    for (int s = 128; s > 0; s >>= 1) {
        if (threadIdx.x < s) {
            s1[threadIdx.x] += s1[threadIdx.x + s];
            s2[threadIdx.x] += s2[threadIdx.x + s];
        }
        __syncthreads();
    }
    if (threadIdx.x == 0) {
        f1[i] = s1[0] + f1b[0];
        f2[i] = s2[0] + f2b[0];
    }
}

// ---------------------------------------------------------------------------
// coefs[i,j] = (i==j) ? 0 : elu(f1[i] + f2[j])
// ---------------------------------------------------------------------------
__global__ __launch_bounds__(256) void coef_kernel(
    const float* __restrict__ f1, const float* __restrict__ f2,
    float* __restrict__ coefs)
{
    const int idx = blockIdx.x * 256 + threadIdx.x;
    if (idx >= N_STOCK * N_STOCK) return;
    const int i = idx / N_STOCK;
    const int j = idx - i * N_STOCK;
    const float x = f1[i] + f2[j];
    const float e = (x > 0.0f) ? x : expm1f(x);
    coefs[idx] = (i == j) ? 0.0f : e;
}

// ---------------------------------------------------------------------------
// out[i,f] = tanh( sum_j coefs[i,j] * seq_s[j,f] * elu(t1[i,f] + t2[j,f]) )
// Block: 128 threads, each owns 4 consecutive f (512 f per block),
// IT=4 rows of i per block. coefs tile staged in LDS; s/t2 float4 loads
// are reused across the IT rows.
// ---------------------------------------------------------------------------
#define IT 4
#define AGG_THREADS 128

__device__ __forceinline__ float elu_f(float x) {
    return (x > 0.0f) ? x : expm1f(x);
}

__global__ __launch_bounds__(AGG_THREADS) void aggregate_kernel(
    const float* __restrict__ coefs, const float* __restrict__ seq_s,
    const float* __restrict__ t1, const float* __restrict__ t2,
    float* __restrict__ out)
{
    __shared__ float cTile[IT][N_STOCK];

    const int i0 = blockIdx.y * IT;
    const int f0 = blockIdx.x * (AGG_THREADS * 4) + threadIdx.x * 4;

    for (int t = threadIdx.x; t < IT * N_STOCK; t += AGG_THREADS) {
        const int ti = t / N_STOCK;
        const int tj = t - ti * N_STOCK;
        const int ir = i0 + ti;
        cTile[ti][tj] = (ir < N_STOCK) ? coefs[ir * N_STOCK + tj] : 0.0f;
    }
    __syncthreads();

    float4 t1v[IT];
    float4 acc[IT];
    #pragma unroll
    for (int t = 0; t < IT; ++t) {
        int ir = i0 + t;
        if (ir >= N_STOCK) ir = N_STOCK - 1;
        t1v[t] = *(const float4*)(t1 + (size_t)ir * F_OUT + f0);
        acc[t] = make_float4(0.0f, 0.0f, 0.0f, 0.0f);
    }

    for (int j = 0; j < N_STOCK; ++j) {
        const float4 sv  = *(const float4*)(seq_s + (size_t)j * F_OUT + f0);
        const float4 t2v = *(const float4*)(t2    + (size_t)j * F_OUT + f0);
        #pragma unroll
        for (int t = 0; t < IT; ++t) {
            const float c = cTile[t][j];
            acc[t].x += c * sv.x * elu_f(t1v[t].x + t2v.x);
            acc[t].y += c * sv.y * elu_f(t1v[t].y + t2v.y);
            acc[t].z += c * sv.z * elu_f(t1v[t].z + t2v.z);
            acc[t].w += c * sv.w * elu_f(t1v[t].w + t2v.w);
        }
    }

    #pragma unroll
    for (int t = 0; t < IT; ++t) {
        const int ir = i0 + t;
        if (ir < N_STOCK) {
            float4 r;
            r.x = tanhf(acc[t].x);
            r.y = tanhf(acc[t].y);
            r.z = tanhf(acc[t].z);
            r.w = tanhf(acc[t].w);
            *(float4*)(out + (size_t)ir * F_OUT + f0) = r;
        }
    }
}

// ---------------------------------------------------------------------------
// Host-side orchestration
// ---------------------------------------------------------------------------
extern "C" void kernel_launch(void* const* d_in, const int* in_sizes, int n_in,
                              void* d_out, int out_size, void* d_ws, size_t ws_size,
                              hipStream_t stream)
{
    const float* input_s = (const float*)d_in[0];
    const float* input_r = (const float*)d_in[1];
    const float* W_r     = (const float*)d_in[2];
    const float* W_s     = (const float*)d_in[3];
    const float* f1_w    = (const float*)d_in[4];
    const float* f1_b    = (const float*)d_in[5];
    const float* f2_w    = (const float*)d_in[6];
    const float* f2_b    = (const float*)d_in[7];
    const float* w1_w    = (const float*)d_in[8];
    const float* w1_b    = (const float*)d_in[9];
    const float* w2_w    = (const float*)d_in[10];
    const float* w2_b    = (const float*)d_in[11];
    float* out = (float*)d_out;

    // workspace layout (bytes)
    const size_t SZ_MAT = (size_t)N_STOCK * F_OUT * sizeof(float);   // 1,622,016
    char* ws = (char*)d_ws;
    float* seq_r = (float*)(ws);
    float* seq_s = (float*)(ws + SZ_MAT);
    float* t1    = (float*)(ws + 2 * SZ_MAT);
    float* t2    = (float*)(ws + 3 * SZ_MAT);
    float* f1    = (float*)(ws + 4 * SZ_MAT);
    float* f2    = (float*)(ws + 4 * SZ_MAT + 1024);
    float* coefs = (float*)(ws + 4 * SZ_MAT + 2048);

    // 1) four WMMA GEMMs: [198,512] x [512,2048]
    const dim3 ggrid(M_TILES, F_OUT / (16 * NT));
    gemm_xwT_wmma<<<ggrid, 32, 0, stream>>>(input_r, W_r,  nullptr, seq_r);
    gemm_xwT_wmma<<<ggrid, 32, 0, stream>>>(input_s, W_s,  nullptr, seq_s);
    gemm_xwT_wmma<<<ggrid, 32, 0, stream>>>(input_s, w1_w, w1_b,    t1);
    gemm_xwT_wmma<<<ggrid, 32, 0, stream>>>(input_s, w2_w, w2_b,    t2);

    // 2) attention logits
    f1f2_kernel<<<N_STOCK, 256, 0, stream>>>(seq_r, f1_w, f2_w, f1_b, f2_b, f1, f2);
    coef_kernel<<<(N_STOCK * N_STOCK + 255) / 256, 256, 0, stream>>>(f1, f2, coefs);

    // 3) fused gated aggregation + tanh
    const dim3 agrid(F_OUT / (AGG_THREADS * 4), (N_STOCK + IT - 1) / IT);
    aggregate_kernel<<<agrid, AGG_THREADS, 0, stream>>>(coefs, seq_s, t1, t2, out);
}